// OT_GNN_layer_34033320853640
// MI455X (gfx1250) — compile-verified
//
#include <hip/hip_runtime.h>
#include <hip/hip_bf16.h>
#include <math.h>

#define NN    5000
#define NE    160000
#define NF    128
#define NH    256
#define NTPL  10
#define NTN   10
#define NCLS  8
#define ALPHA 0.5f
#define CGI   5
#define SKI   30
#define SREG  0.02f

typedef __attribute__((ext_vector_type(2))) float v2f;
typedef __attribute__((ext_vector_type(8))) float v8f;

// ---------------------------------------------------------------------------
// Generic fill
// ---------------------------------------------------------------------------
__global__ void fill_kernel(float* __restrict__ p, float v, int n) {
    int i = blockIdx.x * blockDim.x + threadIdx.x;
    if (i < n) p[i] = v;
}

// ---------------------------------------------------------------------------
// Degree (with self loop) and D^-1/2
// ---------------------------------------------------------------------------
__global__ void deg_edges_kernel(const int* __restrict__ dst, float* __restrict__ deg, int E) {
    int e = blockIdx.x * blockDim.x + threadIdx.x;
    if (e < E) atomicAdd(&deg[dst[e]], 1.0f);
}

__global__ void dinv_kernel(const float* __restrict__ deg, float* __restrict__ dinv, int n) {
    int i = blockIdx.x * blockDim.x + threadIdx.x;
    if (i < n) dinv[i] = rsqrtf(fmaxf(deg[i], 1.0f));
}

__global__ void rowdeg_kernel(const int* __restrict__ src, float* __restrict__ rowdeg, int E) {
    int e = blockIdx.x * blockDim.x + threadIdx.x;
    if (e < E) atomicAdd(&rowdeg[src[e]], 1.0f);
}

// ---------------------------------------------------------------------------
// C = A(MxK) @ B(KxN), fp32, V_WMMA_F32_16X16X4_F32.
// One wave computes a 16x32 output tile: two 16x16 WMMA tiles sharing the A
// fragment (half the A traffic, double WMMA per loaded byte).
// OOB M rows are *clamped* on load (their D rows are never stored), which
// keeps the K-loop free of exec-mask predication.
// A-frag (16x4 f32, 2 VGPR): lanes 0-15 -> M rows, K = {0,1}; lanes 16-31 K={2,3}
// B-frag (4x16 f32, 2 VGPR): vgpr v, lane<16 -> row K=v, col=lane; lane>=16 -> K=v+2
// C/D  (16x16 f32, 8 VGPR): vgpr v, lane<16 -> row v, col=lane; lane>=16 -> row v+8
// ---------------------------------------------------------------------------
__global__ __launch_bounds__(256)
void gemm_wmma_f32(const float* __restrict__ A, const float* __restrict__ B,
                   float* __restrict__ C, int M, int N, int K) {
    const int gt    = blockIdx.x * blockDim.x + threadIdx.x;
    const int wave  = gt >> 5;
    const int lane  = threadIdx.x & 31;
    const int tilesN = N >> 5;              // 32-wide tiles
    const int tilesM = (M + 15) >> 4;
    if (wave >= tilesM * tilesN) return;
    const int m0   = (wave / tilesN) << 4;
    const int n0   = (wave % tilesN) << 5;
    const int half = lane >> 4;             // 0: lanes 0-15, 1: lanes 16-31
    const int l    = lane & 15;
    int arow = m0 + l;
    if (arow >= M) arow = M - 1;            // clamp: garbage flows only to unstored rows
    v8f acc0 = {0.f, 0.f, 0.f, 0.f, 0.f, 0.f, 0.f, 0.f};
    v8f acc1 = {0.f, 0.f, 0.f, 0.f, 0.f, 0.f, 0.f, 0.f};
    for (int k0 = 0; k0 < K; k0 += 4) {
        const int ka = k0 + 2 * half;
        v2f a, b0, b1;
        a.x  = A[arow * K + ka];
        a.y  = A[arow * K + ka + 1];
        b0.x = B[(ka    ) * N + n0 + l];
        b0.y = B[(ka + 1) * N + n0 + l];
        b1.x = B[(ka    ) * N + n0 + 16 + l];
        b1.y = B[(ka + 1) * N + n0 + 16 + l];
        acc0 = __builtin_amdgcn_wmma_f32_16x16x4_f32(
            false, a, false, b0, (short)0, acc0, false, false);
        acc1 = __builtin_amdgcn_wmma_f32_16x16x4_f32(
            false, a, false, b1, (short)0, acc1, false, false);
    }
#pragma unroll
    for (int v = 0; v < 8; ++v) {
        const int row = m0 + v + half * 8;
        if (row < M) {
            C[row * N + n0 + l]      = acc0[v];
            C[row * N + n0 + 16 + l] = acc1[v];
        }
    }
}

// ---------------------------------------------------------------------------
// GCN aggregation: out = D^-1/2 (A+I) D^-1/2 (XW) + b
// ---------------------------------------------------------------------------
__global__ void agg_init_kernel(const float* __restrict__ XW, const float* __restrict__ dinv,
                                const float* __restrict__ bias, float* __restrict__ out,
                                int n, int F) {
    int idx = blockIdx.x * blockDim.x + threadIdx.x;
    if (idx < n * F) {
        int i = idx / F, f = idx - i * F;
        float di = dinv[i];
        out[idx] = XW[idx] * di * di + bias[f];      // self loop + bias
    }
}

__global__ void agg_edges_kernel(const int* __restrict__ src, const int* __restrict__ dst,
                                 const float* __restrict__ XW, const float* __restrict__ dinv,
                                 float* __restrict__ out, int F) {
    int e = blockIdx.x;
    int f = threadIdx.x;
    int s = src[e], d = dst[e];
    float w = dinv[s] * dinv[d];
    atomicAdd(&out[d * F + f], XW[s * F + f] * w);
}

__global__ void relu_kernel(float* __restrict__ x, int n) {
    int i = blockIdx.x * blockDim.x + threadIdx.x;
    if (i < n) x[i] = fmaxf(x[i], 0.0f);
}

// ---------------------------------------------------------------------------
// sx2[i] = sum_f h[i,f]^2
// ---------------------------------------------------------------------------
__global__ void sx2_kernel(const float* __restrict__ H, float* __restrict__ sx2, int n, int F) {
    int i = blockIdx.x * blockDim.x + threadIdx.x;
    if (i < n) {
        float s = 0.f;
        for (int f = 0; f < F; ++f) { float v = H[i * F + f]; s += v * v; }
        sx2[i] = s;
    }
}

// ---------------------------------------------------------------------------
// Feature cost M[t,i,j] = |h_i|^2 + |F2_tj|^2 - 2 <h_i, F2_tj>
// ---------------------------------------------------------------------------
__global__ void mcost_kernel(const float* __restrict__ H, const float* __restrict__ sx2,
                             const float* __restrict__ tplF, float* __restrict__ Mall) {
    int idx = blockIdx.x * blockDim.x + threadIdx.x;
    if (idx >= NTPL * NN * NTN) return;
    int t = idx / (NN * NTN);
    int r = idx - t * (NN * NTN);
    int i = r / NTN;
    int j = r - i * NTN;
    const float* f2 = tplF + ((size_t)t * NTN + j) * NF;
    const float* hi = H + (size_t)i * NF;
    float dot = 0.f, sq = 0.f;
    for (int f = 0; f < NF; ++f) { float a = hi[f], b = f2[f]; dot += a * b; sq += b * b; }
    Mall[idx] = sx2[i] + sq - 2.0f * dot;
}

// ---------------------------------------------------------------------------
// Full FGW solve, one workgroup (256 threads) per template.
// Conditional gradient (5 iters) with log-domain Sinkhorn (30 iters) inner solve.
// ---------------------------------------------------------------------------
__global__ __launch_bounds__(256)
void fgw_kernel(const float* __restrict__ tpl,       // [NTPL,10,10]
                const float* __restrict__ Mall,      // [NTPL,NN,10]
                const float* __restrict__ rowdeg,    // [NN]
                const int* __restrict__ src, const int* __restrict__ dst,
                float* __restrict__ Tmat, float* __restrict__ G,
                float* __restrict__ CT, float* __restrict__ dists) {
    const int t   = blockIdx.x;
    const int tid = threadIdx.x;
    const int nt  = blockDim.x;           // 256

    __shared__ float f_sh[NN];            // 20 KB
    __shared__ float g_sh[NTN];
    __shared__ float c2_sh[NTN * NTN];
    __shared__ float c2sq_sh[NTN];
    __shared__ float pm[256 * NTN];       // 10 KB
    __shared__ float ps[256 * NTN];       // 10 KB
    __shared__ float red[256];
    __shared__ float red2[256];

    const float inv_n = 1.0f / NN, inv_m = 1.0f / NTN;
    const float logp = -logf((float)NN), logq = -logf((float)NTN);
    float*       T   = Tmat + (size_t)t * NN * NTN;
    float*       Gt  = G    + (size_t)t * NN * NTN;
    float*       CTt = CT   + (size_t)t * NN * NTN;
    const float* Mt  = Mall + (size_t)t * NN * NTN;

    if (tid < NTN * NTN) c2_sh[tid] = tpl[t * NTN * NTN + tid];
    __syncthreads();
    if (tid < NTN) {
        float s = 0.f;
        for (int k = 0; k < NTN; ++k) { float v = c2_sh[tid * NTN + k]; s += v * v; }
        c2sq_sh[tid] = s * inv_m;         // ((C2*C2)@q)[j]
    }
    const float invnm = inv_n * inv_m;
    for (int i = tid; i < NN * NTN; i += nt) T[i] = invnm;   // T0 = p q^T
    __threadfence(); __syncthreads();

    for (int cg = 0; cg < CGI; ++cg) {
        // ---- CT = C1 @ T via edge scatter ----
        for (int i = tid; i < NN * NTN; i += nt) CTt[i] = 0.f;
        __threadfence(); __syncthreads();
        for (int e = tid; e < NE; e += nt) {
            int s = src[e], d = dst[e];
#pragma unroll
            for (int k = 0; k < NTN; ++k)
                atomicAdd(&CTt[s * NTN + k], T[d * NTN + k]);
        }
        __threadfence(); __syncthreads();

        // ---- grad = (1-a) M + 2a (constC - CT @ (2 C2)^T), track max|grad| ----
        float lmax = 0.f;
        for (int i = tid; i < NN; i += nt) {
            float ct[NTN];
#pragma unroll
            for (int k = 0; k < NTN; ++k) ct[k] = CTt[i * NTN + k];
            float c1p = rowdeg[i] * inv_n;
#pragma unroll
            for (int j = 0; j < NTN; ++j) {
                float acc = 0.f;
#pragma unroll
                for (int k = 0; k < NTN; ++k) acc += ct[k] * c2_sh[j * NTN + k];
                float tens = c1p + c2sq_sh[j] - 2.0f * acc;
                float grad = (1.0f - ALPHA) * Mt[i * NTN + j] + 2.0f * ALPHA * tens;
                Gt[i * NTN + j] = grad;
                lmax = fmaxf(lmax, fabsf(grad));
            }
        }
        red[tid] = lmax; __syncthreads();
        for (int s = nt >> 1; s > 0; s >>= 1) {
            if (tid < s) red[tid] = fmaxf(red[tid], red[tid + s]);
            __syncthreads();
        }
        const float scale = red[0] + 1e-9f;
        __syncthreads();
        const float fac = -1.0f / (SREG * scale);

        // ---- mlogK in-place (not the writing thread; ordered by barriers above) ----
        for (int i = tid; i < NN * NTN; i += nt) Gt[i] *= fac;
        for (int i = tid; i < NN; i += nt) f_sh[i] = 0.f;
        if (tid < NTN) g_sh[tid] = 0.f;
        __threadfence(); __syncthreads();

        // ---- Sinkhorn, log-domain ----
        for (int it = 0; it < SKI; ++it) {
            // g[j] = logq - LSE_i( K[i,j] + f[i] )
            float mm[NTN], ss[NTN];
#pragma unroll
            for (int j = 0; j < NTN; ++j) { mm[j] = -INFINITY; ss[j] = 0.f; }
            for (int i = tid; i < NN; i += nt) {
                float fi = f_sh[i];
#pragma unroll
                for (int j = 0; j < NTN; ++j) {
                    float v = Gt[i * NTN + j] + fi;
                    if (v > mm[j]) { ss[j] = ss[j] * __expf(mm[j] - v) + 1.0f; mm[j] = v; }
                    else           { ss[j] += __expf(v - mm[j]); }
                }
            }
#pragma unroll
            for (int j = 0; j < NTN; ++j) { pm[tid * NTN + j] = mm[j]; ps[tid * NTN + j] = ss[j]; }
            __syncthreads();
            if (tid < NTN) {
                float M0 = -INFINITY, S0 = 0.f;
                for (int u = 0; u < nt; ++u) {
                    float m2 = pm[u * NTN + tid], s2 = ps[u * NTN + tid];
                    if (m2 > M0) { S0 = S0 * __expf(M0 - m2) + s2; M0 = m2; }
                    else         { S0 += s2 * __expf(m2 - M0); }
                }
                g_sh[tid] = logq - (M0 + logf(S0));
            }
            __syncthreads();
            // f[i] = logp - LSE_j( K[i,j] + g[j] )
            for (int i = tid; i < NN; i += nt) {
                float M1 = -INFINITY, S1 = 0.f;
#pragma unroll
                for (int j = 0; j < NTN; ++j) {
                    float v = Gt[i * NTN + j] + g_sh[j];
                    if (v > M1) { S1 = S1 * __expf(M1 - v) + 1.0f; M1 = v; }
                    else        { S1 += __expf(v - M1); }
                }
                f_sh[i] = logp - (M1 + logf(S1));
            }
            __syncthreads();
        }

        // ---- T += step * (Tnew - T), Tnew = exp(f + K + g) ----
        const float step = 2.0f / (cg + 2.0f);
        for (int i = tid; i < NN; i += nt) {
            float fi = f_sh[i];
#pragma unroll
            for (int j = 0; j < NTN; ++j) {
                int idx = i * NTN + j;
                float Tn = __expf(fi + Gt[idx] + g_sh[j]);
                T[idx] += step * (Tn - T[idx]);
            }
        }
        __threadfence(); __syncthreads();
    }

    // ---- final objective: (1-a) <M,T> + a <tens(T), T> with final T ----
    for (int i = tid; i < NN * NTN; i += nt) CTt[i] = 0.f;
    __threadfence(); __syncthreads();
    for (int e = tid; e < NE; e += nt) {
        int s = src[e], d = dst[e];
#pragma unroll
        for (int k = 0; k < NTN; ++k)
            atomicAdd(&CTt[s * NTN + k], T[d * NTN + k]);
    }
    __threadfence(); __syncthreads();
    float sM = 0.f, sG = 0.f;
    for (int i = tid; i < NN; i += nt) {
        float ct[NTN];
#pragma unroll
        for (int k = 0; k < NTN; ++k) ct[k] = CTt[i * NTN + k];
        float c1p = rowdeg[i] * inv_n;
#pragma unroll
        for (int j = 0; j < NTN; ++j) {
            float acc = 0.f;
#pragma unroll
            for (int k = 0; k < NTN; ++k) acc += ct[k] * c2_sh[j * NTN + k];
            float tens = c1p + c2sq_sh[j] - 2.0f * acc;
            float Tv = T[i * NTN + j];
            sM += Mt[i * NTN + j] * Tv;
            sG += tens * Tv;
        }
    }
    red[tid] = sM; red2[tid] = sG; __syncthreads();
    for (int s = nt >> 1; s > 0; s >>= 1) {
        if (tid < s) { red[tid] += red[tid + s]; red2[tid] += red2[tid + s]; }
        __syncthreads();
    }
    if (tid == 0) dists[t] = (1.0f - ALPHA) * red[0] + ALPHA * red2[0];
}

// ---------------------------------------------------------------------------
// out[c] = dists @ Wlin + blin
// ---------------------------------------------------------------------------
__global__ void head_kernel(const float* __restrict__ dists, const float* __restrict__ Wlin,
                            const float* __restrict__ blin, float* __restrict__ out) {
    int c = threadIdx.x;
    if (c < NCLS) {
        float s = blin[c];
        for (int t = 0; t < NTPL; ++t) s += dists[t] * Wlin[t * NCLS + c];
        out[c] = s;
    }
}

// ---------------------------------------------------------------------------
extern "C" void kernel_launch(void* const* d_in, const int* in_sizes, int n_in,
                              void* d_out, int out_size, void* d_ws, size_t ws_size,
                              hipStream_t stream) {
    const float* x    = (const float*)d_in[0];   // [NN, NF]
    const int*   ei   = (const int*)  d_in[1];   // [2, NE]
    const float* W1   = (const float*)d_in[2];   // [NF, NH]
    const float* b1   = (const float*)d_in[3];   // [NH]
    const float* W2   = (const float*)d_in[4];   // [NH, NF]
    const float* b2   = (const float*)d_in[5];   // [NF]
    const float* tpl  = (const float*)d_in[6];   // [NTPL, NTN, NTN]
    const float* tplF = (const float*)d_in[7];   // [NTPL, NTN, NF]
    const float* Wlin = (const float*)d_in[8];   // [NTPL, NCLS]
    const float* blin = (const float*)d_in[9];   // [NCLS]
    float* out = (float*)d_out;

    const int* src = ei;
    const int* dst = ei + NE;

    // workspace carve-up (floats)
    float* w = (float*)d_ws;
    float* deg    = w; w += NN;
    float* dinv   = w; w += NN;
    float* rowdeg = w; w += NN;
    float* sx2    = w; w += NN;
    float* XW1    = w; w += (size_t)NN * NH;
    float* H1     = w; w += (size_t)NN * NH;
    float* XW2    = w; w += (size_t)NN * NF;
    float* H      = w; w += (size_t)NN * NF;
    float* Mall   = w; w += (size_t)NTPL * NN * NTN;
    float* Tmat   = w; w += (size_t)NTPL * NN * NTN;
    float* G      = w; w += (size_t)NTPL * NN * NTN;
    float* CT     = w; w += (size_t)NTPL * NN * NTN;
    float* dists  = w; w += NTPL;

    const int B = 256;

    // degrees + D^-1/2 (deg starts at 1.0 for the self loop)
    fill_kernel<<<(NN + B - 1) / B, B, 0, stream>>>(deg, 1.0f, NN);
    fill_kernel<<<(NN + B - 1) / B, B, 0, stream>>>(rowdeg, 0.0f, NN);
    deg_edges_kernel<<<(NE + B - 1) / B, B, 0, stream>>>(dst, deg, NE);
    rowdeg_kernel<<<(NE + B - 1) / B, B, 0, stream>>>(src, rowdeg, NE);
    dinv_kernel<<<(NN + B - 1) / B, B, 0, stream>>>(deg, dinv, NN);

    const int tilesM = (NN + 15) / 16;

    // ---- layer 1: H1 = relu( Dhat (x @ W1) + b1 ) ----
    {
        int waves = tilesM * (NH / 32);
        int blks = (waves * 32 + B - 1) / B;
        gemm_wmma_f32<<<blks, B, 0, stream>>>(x, W1, XW1, NN, NH, NF);
    }
    agg_init_kernel<<<(NN * NH + B - 1) / B, B, 0, stream>>>(XW1, dinv, b1, H1, NN, NH);
    agg_edges_kernel<<<NE, NH, 0, stream>>>(src, dst, XW1, dinv, H1, NH);
    relu_kernel<<<(NN * NH + B - 1) / B, B, 0, stream>>>(H1, NN * NH);

    // ---- layer 2: H = Dhat (H1 @ W2) + b2 ----
    {
        int waves = tilesM * (NF / 32);
        int blks = (waves * 32 + B - 1) / B;
        gemm_wmma_f32<<<blks, B, 0, stream>>>(H1, W2, XW2, NN, NF, NH);
    }
    agg_init_kernel<<<(NN * NF + B - 1) / B, B, 0, stream>>>(XW2, dinv, b2, H, NN, NF);
    agg_edges_kernel<<<NE, NF, 0, stream>>>(src, dst, XW2, dinv, H, NF);

    // ---- feature costs ----
    sx2_kernel<<<(NN + B - 1) / B, B, 0, stream>>>(H, sx2, NN, NF);
    mcost_kernel<<<(NTPL * NN * NTN + B - 1) / B, B, 0, stream>>>(H, sx2, tplF, Mall);

    // ---- FGW per template (persistent block each) ----
    fgw_kernel<<<NTPL, B, 0, stream>>>(tpl, Mall, rowdeg, src, dst, Tmat, G, CT, dists);

    // ---- classifier head ----
    head_kernel<<<1, 64, 0, stream>>>(dists, Wlin, blin, out);
}